// GAP_78030965834371
// MI455X (gfx1250) — compile-verified
//
#include <hip/hip_runtime.h>

typedef __attribute__((ext_vector_type(2))) float v2f;
typedef __attribute__((ext_vector_type(8))) float v8f;

#define E_SEGS 131072
#define D_IN   128
#define D_HID  512
#define D_OUT  128

// ---------------------------------------------------------------------------
// Weight repack: K-pair interleave so a lane's WMMA B-fragment (W[k][n],
// W[k+1][n]) becomes one aligned 8-byte load (global_load_b64), with 16
// consecutive lanes forming a coalesced 128B segment.
//   Wp[k/2][n] = { W[k][n], W[k+1][n] }
// ---------------------------------------------------------------------------
__global__ __launch_bounds__(256) void pack_pairs_kernel(
    const float* __restrict__ W, v2f* __restrict__ Wp, int ncols, int total_pairs) {
  int i = blockIdx.x * blockDim.x + threadIdx.x;   // i = kp * ncols + n
  if (i >= total_pairs) return;
  int kp = i / ncols;
  int n  = i - kp * ncols;
  v2f p;
  p[0] = W[(2 * kp) * ncols + n];
  p[1] = W[(2 * kp + 1) * ncols + n];
  Wp[i] = p;
}

// ---------------------------------------------------------------------------
// Phase 1: segmented sum + counts via L2-resident f32 atomics.
// One 32-lane wave per incidence entry; each lane handles 4 columns (float4).
// ---------------------------------------------------------------------------
__global__ __launch_bounds__(256) void seg_scatter_kernel(
    const float* __restrict__ x,
    const int* __restrict__ hidx,   // [2, NNZ] int32; row 1 = segment ids
    float* __restrict__ sums,       // [E, 128] (pre-zeroed; aliases d_out)
    float* __restrict__ cnt,        // [E] (pre-zeroed)
    int nnz) {
  int t = blockIdx.x * blockDim.x + threadIdx.x;
  int e = t >> 5;
  int lane = t & 31;
  if (e >= nnz) return;
  int seg = hidx[nnz + e];
  float4 v = *((const float4*)(x + (size_t)e * D_IN) + lane);
  float* dst = sums + (size_t)seg * D_IN + lane * 4;
  atomicAdd(dst + 0, v.x);
  atomicAdd(dst + 1, v.y);
  atomicAdd(dst + 2, v.z);
  atomicAdd(dst + 3, v.w);
  if (lane == 0) atomicAdd(cnt + seg, 1.0f);
}

// ---------------------------------------------------------------------------
// Phase 2: fused  out = relu(mean @ W1 + b1) @ W2 + b2  via f32 WMMA.
// One wave per 16-row tile; mean A-fragments register-resident (32 x v2f);
// per 16-wide hidden tile: 32 WMMAs (K=128), bias+ReLU, 1KB LDS C->A
// transpose, 32 WMMAs into the 8 output accumulator tiles (N=128).
// With packed weights: 64 b64 loads per j vs 64 WMMAs (1:1 issue ratio).
// ---------------------------------------------------------------------------
__global__ __launch_bounds__(128) void mlp_wmma_kernel(
    const float* sums,              // [E, 128]  (aliases out - no restrict!)
    const float* __restrict__ cnt,  // [E]
    const v2f* __restrict__ W1p,    // [64, 512] K-pair packed
    const float* __restrict__ b1,   // [512]
    const v2f* __restrict__ W2p,    // [256, 128] K-pair packed
    const float* __restrict__ b2,   // [128]
    float* out) {                   // [E, 128]  (aliases sums - no restrict!)
  __shared__ float hbuf[4][16 * 18];          // stride 18: bank-conflict-free, 8B aligned
  const int wave = threadIdx.x >> 5;
  const int lane = threadIdx.x & 31;
  const int hl = lane >> 4;                   // lane half: selects K sub-pair
  const int m = lane & 15;                    // row within tile / N within tile
  const int Mbase = (blockIdx.x * 4 + wave) * 16;

  // Per-row mean scale 1/max(cnt,1)
  const float rinv = 1.0f / fmaxf(cnt[Mbase + m], 1.0f);

  // Load full 16x128 mean tile as 32 A-fragments.
  // A layout: lanes 0-15 -> K = 4kc+{0,1}; lanes 16-31 -> K = 4kc+{2,3}.
  v2f A[32];
  const float* srow = sums + (size_t)(Mbase + m) * D_IN;
  #pragma unroll
  for (int kc = 0; kc < 32; ++kc) {
    v2f t = *(const v2f*)(srow + kc * 4 + hl * 2);   // 8B aligned
    A[kc] = t * rinv;
  }

  v8f Oacc[8];
  #pragma unroll
  for (int n = 0; n < 8; ++n) Oacc[n] = {};

  float* hrow = hbuf[wave];

  for (int j = 0; j < 32; ++j) {
    // ---- GEMM1: htile[16x16] = meanTile[16x128] @ W1[:, j*16 .. j*16+15]
    v8f H = {};
    #pragma unroll
    for (int kc = 0; kc < 32; ++kc) {
      v2f B = W1p[(size_t)(2 * kc + hl) * D_HID + (j * 16 + m)];  // one b64 load
      H = __builtin_amdgcn_wmma_f32_16x16x4_f32(false, A[kc], false, B,
                                                (short)0, H, false, false);
    }
    // bias + ReLU in C-layout (every VGPR holds N = m)
    {
      const float bb = b1[j * 16 + m];
      #pragma unroll
      for (int r = 0; r < 8; ++r) H[r] = fmaxf(H[r] + bb, 0.0f);
    }
    // ---- C-layout -> A-layout via LDS (same-wave producer/consumer)
    #pragma unroll
    for (int r = 0; r < 8; ++r)
      hrow[(r + hl * 8) * 18 + m] = H[r];     // h[row][col], row-major stride 18
    asm volatile("s_wait_dscnt 0" ::: "memory");
    // ---- GEMM2: Oacc[n] += htile[16x16] @ W2[j*16 .. , n*16 ..]
    #pragma unroll
    for (int kk = 0; kk < 4; ++kk) {
      const int k0 = kk * 4 + hl * 2;
      v2f Ah = *(const v2f*)(hrow + m * 18 + k0);    // 72m + 4k0 : 8B aligned
      #pragma unroll
      for (int n = 0; n < 8; ++n) {
        // K-pair index: (j*16 + k0)/2 = j*8 + 2*kk + hl
        v2f Bh = W2p[(size_t)(j * 8 + 2 * kk + hl) * D_OUT + (n * 16 + m)];
        Oacc[n] = __builtin_amdgcn_wmma_f32_16x16x4_f32(false, Ah, false, Bh,
                                                        (short)0, Oacc[n],
                                                        false, false);
      }
    }
    asm volatile("s_wait_dscnt 0" ::: "memory");    // fence before hrow reuse
  }

  // ---- Epilogue: + b2, store C-layout (VGPR r -> row r + hl*8, col n*16+m)
  #pragma unroll
  for (int n = 0; n < 8; ++n) {
    const float bb = b2[n * 16 + m];
    #pragma unroll
    for (int r = 0; r < 8; ++r)
      out[(size_t)(Mbase + r + hl * 8) * D_OUT + (n * 16 + m)] = Oacc[n][r] + bb;
  }
}

// ---------------------------------------------------------------------------
extern "C" void kernel_launch(void* const* d_in, const int* in_sizes, int n_in,
                              void* d_out, int out_size, void* d_ws, size_t ws_size,
                              hipStream_t stream) {
  const float* x    = (const float*)d_in[0];
  const int*   hidx = (const int*)d_in[1];
  const float* W1   = (const float*)d_in[2];
  const float* b1   = (const float*)d_in[3];
  const float* W2   = (const float*)d_in[4];
  const float* b2   = (const float*)d_in[5];
  float* out = (float*)d_out;

  // Workspace layout: cnt [E] f32 | W1p [64*512] v2f | W2p [256*128] v2f
  char* ws = (char*)d_ws;
  float* cnt = (float*)ws;                                   // 512 KB
  v2f* W1p = (v2f*)(ws + (size_t)E_SEGS * sizeof(float));    // 256 KB
  v2f* W2p = (v2f*)(ws + (size_t)E_SEGS * sizeof(float) + (D_IN / 2) * D_HID * sizeof(v2f));

  const int nnz = in_sizes[1] / 2;  // hyperedge_index is [2, NNZ]

  // Zero accumulation buffers (d_out doubles as the sums scratch).
  hipMemsetAsync(d_out, 0, (size_t)E_SEGS * D_IN * sizeof(float), stream);
  hipMemsetAsync(cnt, 0, (size_t)E_SEGS * sizeof(float), stream);

  // Repack weights into K-pair-interleaved float2 (one b64 per B-fragment).
  {
    const int p1 = (D_IN / 2) * D_HID;   // 32768 pairs
    const int p2 = (D_HID / 2) * D_OUT;  // 32768 pairs
    pack_pairs_kernel<<<(p1 + 255) / 256, 256, 0, stream>>>(W1, W1p, D_HID, p1);
    pack_pairs_kernel<<<(p2 + 255) / 256, 256, 0, stream>>>(W2, W2p, D_OUT, p2);
  }

  // Phase 1: segmented sum + counts.
  {
    const int threads = 256;
    const long long total = (long long)nnz * 32;
    const int blocks = (int)((total + threads - 1) / threads);
    seg_scatter_kernel<<<blocks, threads, 0, stream>>>(x, hidx, out, cnt, nnz);
  }

  // Phase 2: fused mean-normalize + MLP with f32 WMMA.
  // 4 waves/block, one 16-row tile per wave -> 64 rows/block.
  mlp_wmma_kernel<<<E_SEGS / 64, 128, 0, stream>>>(out, cnt, W1p, b1, W2p, b2, out);
}